// ComplexLinearRNN_5866925326617
// MI455X (gfx1250) — compile-verified
//
#include <hip/hip_runtime.h>

// ---------------------------------------------------------------------------
// ComplexLinearRNN on MI455X (gfx1250, wave32, WMMA, TDM)
//   h_t = xp_t + h_{t-1} * M,  M = i * Wh^T   (row-vector convention)
// Chunked parallel scan: T=4096 = NC(64) chunks * L(64) steps.
//   K0: build M = i*Wh^T
//   K1: M^64 via 6 repeated squarings (complex 128x128 WMMA GEMMs)
//   K2: phase1 - per-chunk local scan from 0 -> chunk finals f_j   (parallel)
//   K3: phase2 - carries c_{j+1} = c_j*M^64 + f_j                  (serial, 1 WG)
//   K4: phase3 - per-chunk scan from c_j, fused in/out projections (parallel)
// All matmuls use V_WMMA_F32_16X16X4_F32; B=16 is exactly the WMMA M dim.
// M/Wi B-fragments are register-resident across the serial loop; M is staged
// into padded LDS by the Tensor Data Mover (tensor_load_to_lds, 6-arg form).
// ---------------------------------------------------------------------------

typedef __attribute__((ext_vector_type(2))) float v2f;
typedef __attribute__((ext_vector_type(8))) float v8f;
typedef __attribute__((ext_vector_type(4))) unsigned int v4u;
typedef __attribute__((ext_vector_type(4))) int v4i;
typedef __attribute__((ext_vector_type(8))) int v8i;

#define BB   16
#define TT   4096
#define DIN  64
#define DOUT 64
#define HH   128
#define LCH  64   // chunk length
#define NC   64   // number of chunks

// LDS strides chosen so the two 16-lane half-wave access sets hit disjoint banks
#define BSTR 136  // B-operand tiles, 128 wide (bank = 8k+n vs 8k+16+n)
#define OSTR 72   // B-operand tiles, 64 wide
#define ASTR 132  // A-operand h/c tiles (bank = 4m+kb, all distinct)

#if __has_builtin(__builtin_amdgcn_tensor_load_to_lds) && \
    __has_builtin(__builtin_amdgcn_s_wait_tensorcnt)
#define HAVE_TDM 1
#else
#define HAVE_TDM 0
#endif

__device__ __forceinline__ v8f wmma4(v2f a, v2f b, v8f c) {
  // f32 WMMA: NEG on A/B must be 0; c_mod=0; no reuse hints.
  return __builtin_amdgcn_wmma_f32_16x16x4_f32(
      false, a, false, b, (short)0, c, false, false);
}

// ---- fragment loaders (ISA 7.12.2 wave32 layouts) --------------------------
// A (16x4 f32): lanes 0-15 rows M=0..15 K={k0,k0+1}; lanes 16-31 K={k0+2,k0+3}
__device__ __forceinline__ v2f ldA_g(const float* A, int m0, int k0, int lda, int lane) {
  const float* p = A + (size_t)(m0 + (lane & 15)) * lda + (k0 + ((lane >> 4) << 1));
  return v2f{p[0], p[1]};
}
__device__ __forceinline__ v2f ldA_x(const float* x, int t, int k0, int lane) {
  // x is [B][T][DIN]; WMMA row m = batch index
  const float* p = x + ((size_t)(lane & 15) * TT + t) * DIN + (k0 + ((lane >> 4) << 1));
  return v2f{p[0], p[1]};
}
__device__ __forceinline__ v2f ldA_s(const float* hs, int k0, int lds, int lane) {
  const float* p = hs + (lane & 15) * lds + (k0 + ((lane >> 4) << 1));
  return v2f{p[0], p[1]};
}
// B (4x16 f32): N across lanes within a VGPR, K rows split across half-waves
__device__ __forceinline__ v2f ldB_g(const float* Bm, int k0, int n0, int ldb, int lane) {
  int n = n0 + (lane & 15);
  int kb = k0 + ((lane >> 4) << 1);
  return v2f{Bm[(size_t)kb * ldb + n], Bm[(size_t)(kb + 1) * ldb + n]};
}
__device__ __forceinline__ v2f ldB_s(const float* Bs, int k0, int n0, int ldb, int lane) {
  int n = n0 + (lane & 15);
  int kb = k0 + ((lane >> 4) << 1);
  return v2f{Bs[kb * ldb + n], Bs[(kb + 1) * ldb + n]};
}
// C/D (16x16 f32): VGPR v -> row v + 8*half, col = lane&15 (+n0)
__device__ __forceinline__ void stC_g(float* Cm, const v8f& c, int m0, int n0, int ldc, int lane) {
  int n = n0 + (lane & 15);
  int rb = m0 + ((lane >> 4) << 3);
#pragma unroll
  for (int v = 0; v < 8; ++v) Cm[(size_t)(rb + v) * ldc + n] = c[v];
}
__device__ __forceinline__ void stC_s(float* Cs, const v8f& c, int ldc, int lane) {
  int n = lane & 15;
  int rb = (lane >> 4) << 3;
#pragma unroll
  for (int v = 0; v < 8; ++v) Cs[(rb + v) * ldc + n] = c[v];
}

// ---- Tensor Data Mover: 128x128 f32 tile, global row stride 128 (compact),
// LDS destination padded +8 DWORDs after every 128 DWORDs -> row stride 136.
__device__ __forceinline__ void tdm_load_M(const float* g, float* lds_dst) {
#if HAVE_TDM
  unsigned long long ga = (unsigned long long)g;
  v4u g0;
  g0[0] = 1u;                                   // count=1, user mode, no gather
  g0[1] = (unsigned int)(unsigned long long)lds_dst;  // lds_addr (byte offset)
  g0[2] = (unsigned int)ga;                     // global_addr[31:0]
  g0[3] = (unsigned int)((ga >> 32) & 0x01FFFFFFu) | (2u << 30); // addr[56:32]|type=2
  v8i g1;
  g1[0] = (2 << 16)      // data_size = 4 bytes
        | (1 << 20)      // pad_enable
        | (6 << 22)      // pad_interval: pad after 128 DWORDs
        | (7 << 25);     // pad_amount: 8 DWORDs
  g1[1] = HH << 16;      // tensor_dim0[15:0] = 128 (bits 63:48)
  g1[2] = HH << 16;      // tensor_dim1[15:0] = 128 (bits 95:80)
  g1[3] = HH << 16;      // tile_dim0 = 128 (bits 127:112)
  g1[4] = HH;            // tile_dim1 = 128, tile_dim2 = 0
  g1[5] = HH;            // tensor_dim0_stride = 128 elements
  g1[6] = 0;
  g1[7] = 0;
  v4i z4 = {0, 0, 0, 0};
  v8i z8 = {0, 0, 0, 0, 0, 0, 0, 0};
  // 6-arg form (clang-23 / therock headers): groups 0..3 + extra group + cpol
  __builtin_amdgcn_tensor_load_to_lds(g0, g1, z4, z4, z8, 0);
#else
  (void)g; (void)lds_dst;
#endif
}

// ---------------------------------------------------------------------------
// K0: M_re[k][n] = -Wh_im[n][k]; M_im[k][n] = Wh_re[n][k]   (M = i*Wh^T)
__global__ void build_M_kernel(const float* __restrict__ Wh_re,
                               const float* __restrict__ Wh_im,
                               float* __restrict__ M) {
  int i = blockIdx.x * 256 + threadIdx.x;
  if (i < HH * HH) {
    int k = i >> 7, n = i & 127;
    M[i]           = -Wh_im[n * HH + k];
    M[HH * HH + i] =  Wh_re[n * HH + k];
  }
}

// K1: C = A*A (complex 128x128). 64 tiles, one per wave; 8 WGs x 8 waves.
__global__ void mat_square_kernel(const float* __restrict__ A, float* __restrict__ C) {
  const float* Are = A;
  const float* Aim = A + HH * HH;
  int lane = threadIdx.x & 31;
  int tile = blockIdx.x * 8 + (threadIdx.x >> 5);
  int m0 = (tile >> 3) << 4, n0 = (tile & 7) << 4;
  v8f cP = {}, cN = {}, cI = {};
#pragma unroll 4
  for (int ks = 0; ks < 32; ++ks) {
    int k0 = ks << 2;
    v2f ar = ldA_g(Are, m0, k0, HH, lane);
    v2f ai = ldA_g(Aim, m0, k0, HH, lane);
    v2f br = ldB_g(Are, k0, n0, HH, lane);
    v2f bi = ldB_g(Aim, k0, n0, HH, lane);
    cP = wmma4(ar, br, cP);   // +Are*Are
    cN = wmma4(ai, bi, cN);   // -Aim*Aim (sign applied at the end)
    cI = wmma4(ar, bi, cI);
    cI = wmma4(ai, br, cI);
  }
  v8f cre;
#pragma unroll
  for (int v = 0; v < 8; ++v) cre[v] = cP[v] - cN[v];
  stC_g(C,           cre, m0, n0, HH, lane);
  stC_g(C + HH * HH, cI,  m0, n0, HH, lane);
}

// K2/K4: per-chunk scan. PHASE3=false: h0=0, emit chunk final f_j.
//        PHASE3=true : h0=c_j, fuse out_t = h_t @ Wo^T -> d_out.
template <bool PHASE3>
__global__ void chunk_kernel(const float* __restrict__ x_re, const float* __restrict__ x_im,
                             const float* __restrict__ Wi_re, const float* __restrict__ Wi_im,
                             const float* __restrict__ Wo_re, const float* __restrict__ Wo_im,
                             const float* __restrict__ M,     // i*Wh^T, re|im planar
                             const float* __restrict__ Cst,   // carries (phase3)
                             float* __restrict__ Fout,        // chunk finals (phase1)
                             float* __restrict__ out) {
  extern __shared__ float sm[];
  float* Ms_re  = sm;                       // [128][BSTR]
  float* Ms_im  = Ms_re + HH * BSTR;
  float* Wis_re = Ms_im + HH * BSTR;        // Wi^T [64][BSTR]
  float* Wis_im = Wis_re + DIN * BSTR;
  float* hbuf   = Wis_im + DIN * BSTR;      // [2 buf][2 cmp][16][ASTR]
  float* Wos_re = hbuf + 2 * 2 * BB * ASTR; // Wo^T [128][OSTR] (phase3 only)
  float* Wos_im = Wos_re + HH * OSTR;

  const int tid = threadIdx.x, lane = tid & 31, wave = tid >> 5;
  const int j = blockIdx.x;

#if HAVE_TDM
  if (wave == 0) {             // one DMA per plane; TENSORcnt tracks completion
    tdm_load_M(M,           Ms_re);
    tdm_load_M(M + HH * HH, Ms_im);
  }
#else
  for (int i = tid; i < HH * HH; i += 256) {
    int k = i >> 7, n = i & 127;
    Ms_re[k * BSTR + n] = M[i];
    Ms_im[k * BSTR + n] = M[HH * HH + i];
  }
#endif
  for (int i = tid; i < DIN * HH; i += 256) {   // Wi^T[k][n] = Wi[n][k]
    int k = i >> 7, n = i & 127;
    Wis_re[k * BSTR + n] = Wi_re[n * DIN + k];
    Wis_im[k * BSTR + n] = Wi_im[n * DIN + k];
  }
  if (PHASE3) {
    for (int i = tid; i < HH * DOUT; i += 256) { // Wo^T[k][n] = Wo[n][k]
      int k = i >> 6, n = i & 63;
      Wos_re[k * OSTR + n] = Wo_re[n * HH + k];
      Wos_im[k * OSTR + n] = Wo_im[n * HH + k];
    }
  }
  for (int i = tid; i < BB * ASTR; i += 256) {  // init h buffer 0
    int m = i / ASTR, n = i % ASTR;
    float vr = 0.f, vi = 0.f;
    if (PHASE3 && n < HH) {
      vr = Cst[j * 4096 + m * HH + n];
      vi = Cst[j * 4096 + 2048 + m * HH + n];
    }
    hbuf[i] = vr;
    hbuf[BB * ASTR + i] = vi;
  }
#if HAVE_TDM
  if (wave == 0) __builtin_amdgcn_s_wait_tensorcnt(0);
#endif
  __syncthreads();

  const int n0 = wave << 4;  // this wave's H tile

  // Register-resident stationary B fragments (stay live across the serial loop)
  v2f mbr[32], mbi[32];
#pragma unroll
  for (int ks = 0; ks < 32; ++ks) {
    mbr[ks] = ldB_s(Ms_re, ks << 2, n0, BSTR, lane);
    mbi[ks] = ldB_s(Ms_im, ks << 2, n0, BSTR, lane);
  }
  v2f wir[16], wii[16];
#pragma unroll
  for (int ks = 0; ks < 16; ++ks) {
    wir[ks] = ldB_s(Wis_re, ks << 2, n0, BSTR, lane);
    wii[ks] = ldB_s(Wis_im, ks << 2, n0, BSTR, lane);
  }

#pragma unroll 1
  for (int s = 0; s < LCH; ++s) {
    const int t = j * LCH + s;
    const int cur = s & 1, nxt = cur ^ 1;
    const float* hr = hbuf + (cur * 2 + 0) * BB * ASTR;
    const float* hi = hbuf + (cur * 2 + 1) * BB * ASTR;
    float* hnr = hbuf + (nxt * 2 + 0) * BB * ASTR;
    float* hni = hbuf + (nxt * 2 + 1) * BB * ASTR;

    if (s + 1 < LCH) {  // prefetch next timestep's x rows (global_prefetch_b8)
      __builtin_prefetch(x_re + ((size_t)(lane & 15) * TT + (t + 1)) * DIN, 0, 1);
      __builtin_prefetch(x_im + ((size_t)(lane & 15) * TT + (t + 1)) * DIN, 0, 1);
    }

    // Re = P - N;  Im = I  (sign-free accumulation, subtraction once per step)
    v8f aP = {}, aN = {}, aI = {};
#pragma unroll
    for (int ks = 0; ks < 16; ++ks) {  // xp_t = x_t @ Wi^T  (K=64)
      int k0 = ks << 2;
      v2f xr = ldA_x(x_re, t, k0, lane);
      v2f xi = ldA_x(x_im, t, k0, lane);
      aP = wmma4(xr, wir[ks], aP);
      aN = wmma4(xi, wii[ks], aN);
      aI = wmma4(xr, wii[ks], aI);
      aI = wmma4(xi, wir[ks], aI);
    }
#pragma unroll
    for (int ks = 0; ks < 32; ++ks) {  // + h_{t-1} @ M  (K=128)
      int k0 = ks << 2;
      v2f ar = ldA_s(hr, k0, ASTR, lane);
      v2f ai = ldA_s(hi, k0, ASTR, lane);
      aP = wmma4(ar, mbr[ks], aP);
      aN = wmma4(ai, mbi[ks], aN);
      aI = wmma4(ar, mbi[ks], aI);
      aI = wmma4(ai, mbr[ks], aI);
    }
    v8f aRe;
#pragma unroll
    for (int v = 0; v < 8; ++v) aRe[v] = aP[v] - aN[v];
    stC_s(hnr + n0, aRe, ASTR, lane);
    stC_s(hni + n0, aI,  ASTR, lane);
    __syncthreads();  // publish h_t (also orders next iter's writes vs our reads)

    if (PHASE3) {  // out_t = h_t @ Wo^T; waves 0-3: re tiles, 4-7: im tiles
      const bool isIm = wave >= 4;
      const int no = (wave & 3) << 4;
      v8f oP = {}, oN = {};
#pragma unroll 4
      for (int ks = 0; ks < 32; ++ks) {
        int k0 = ks << 2;
        v2f ar = ldA_s(hnr, k0, ASTR, lane);
        v2f ai = ldA_s(hni, k0, ASTR, lane);
        if (!isIm) {
          oP = wmma4(ar, ldB_s(Wos_re, k0, no, OSTR, lane), oP);
          oN = wmma4(ai, ldB_s(Wos_im, k0, no, OSTR, lane), oN);
        } else {
          oP = wmma4(ar, ldB_s(Wos_im, k0, no, OSTR, lane), oP);
          oP = wmma4(ai, ldB_s(Wos_re, k0, no, OSTR, lane), oP);
        }
      }
      const int n = no + (lane & 15);
      const int rb = (lane >> 4) << 3;
#pragma unroll
      for (int v = 0; v < 8; ++v) {  // complex64 -> interleaved (re, im) floats
        size_t idx = (((size_t)(rb + v) * TT + t) * DOUT + n) * 2 + (isIm ? 1 : 0);
        out[idx] = isIm ? oP[v] : (oP[v] - oN[v]);
      }
    }
  }

  if (!PHASE3) {  // LCH even -> final state is in buffer 0
    const float* fr = hbuf;
    const float* fi = hbuf + BB * ASTR;
    for (int i = tid; i < BB * HH; i += 256) {
      int m = i >> 7, n = i & 127;
      Fout[j * 4096 + i]        = fr[m * ASTR + n];
      Fout[j * 4096 + 2048 + i] = fi[m * ASTR + n];
    }
  }
}

// K3: carries. c_0 = 0; store c_j; c_{j+1} = c_j * M^L + f_j. Single WG.
__global__ void carry_kernel(const float* __restrict__ Mp, const float* __restrict__ F,
                             float* __restrict__ Cst) {
  extern __shared__ float sm[];
  float* Ms_re = sm;
  float* Ms_im = Ms_re + HH * BSTR;
  float* cbuf  = Ms_im + HH * BSTR;  // [2][2][16][ASTR]
  const int tid = threadIdx.x, lane = tid & 31, wave = tid >> 5;

#if HAVE_TDM
  if (wave == 0) {
    tdm_load_M(Mp,           Ms_re);
    tdm_load_M(Mp + HH * HH, Ms_im);
  }
#else
  for (int i = tid; i < HH * HH; i += 256) {
    int k = i >> 7, n = i & 127;
    Ms_re[k * BSTR + n] = Mp[i];
    Ms_im[k * BSTR + n] = Mp[HH * HH + i];
  }
#endif
  for (int i = tid; i < BB * ASTR; i += 256) {
    cbuf[i] = 0.f;
    cbuf[BB * ASTR + i] = 0.f;
  }
#if HAVE_TDM
  if (wave == 0) __builtin_amdgcn_s_wait_tensorcnt(0);
#endif
  __syncthreads();

  const int n0 = wave << 4;
  v2f mbr[32], mbi[32];
#pragma unroll
  for (int ks = 0; ks < 32; ++ks) {
    mbr[ks] = ldB_s(Ms_re, ks << 2, n0, BSTR, lane);
    mbi[ks] = ldB_s(Ms_im, ks << 2, n0, BSTR, lane);
  }

#pragma unroll 1
  for (int jj = 0; jj < NC; ++jj) {
    const int cur = jj & 1, nxt = cur ^ 1;
    const float* cr = cbuf + (cur * 2 + 0) * BB * ASTR;
    const float* ci = cbuf + (cur * 2 + 1) * BB * ASTR;
    float* cnr = cbuf + (nxt * 2 + 0) * BB * ASTR;
    float* cni = cbuf + (nxt * 2 + 1) * BB * ASTR;

    for (int i = tid; i < BB * HH; i += 256) {  // emit c_j
      int m = i >> 7, n = i & 127;
      Cst[jj * 4096 + i]        = cr[m * ASTR + n];
      Cst[jj * 4096 + 2048 + i] = ci[m * ASTR + n];
    }

    v8f aP = {}, aN = {}, aI = {};
#pragma unroll
    for (int ks = 0; ks < 32; ++ks) {
      int k0 = ks << 2;
      v2f ar = ldA_s(cr, k0, ASTR, lane);
      v2f ai = ldA_s(ci, k0, ASTR, lane);
      aP = wmma4(ar, mbr[ks], aP);
      aN = wmma4(ai, mbi[ks], aN);
      aI = wmma4(ar, mbi[ks], aI);
      aI = wmma4(ai, mbr[ks], aI);
    }
    {  // Re = P - N + f_re;  Im = I + f_im
      const int n = n0 + (lane & 15);
      const int rb = (lane >> 4) << 3;
#pragma unroll
      for (int v = 0; v < 8; ++v) {
        aP[v] = aP[v] - aN[v] + F[jj * 4096 + (rb + v) * HH + n];
        aI[v] += F[jj * 4096 + 2048 + (rb + v) * HH + n];
      }
    }
    stC_s(cnr + n0, aP, ASTR, lane);
    stC_s(cni + n0, aI, ASTR, lane);
    __syncthreads();
  }
}

// ---------------------------------------------------------------------------
extern "C" void kernel_launch(void* const* d_in, const int* in_sizes, int n_in,
                              void* d_out, int out_size, void* d_ws, size_t ws_size,
                              hipStream_t stream) {
  const float* x_re  = (const float*)d_in[0];
  const float* x_im  = (const float*)d_in[1];
  const float* Wi_re = (const float*)d_in[2];
  const float* Wi_im = (const float*)d_in[3];
  const float* Wh_re = (const float*)d_in[4];
  const float* Wh_im = (const float*)d_in[5];
  const float* Wo_re = (const float*)d_in[6];
  const float* Wo_im = (const float*)d_in[7];
  float* out = (float*)d_out;

  // workspace layout (floats): Mbase | P0 | P1 | F[NC] | Cst[NC]  (~2.5 MB)
  float* ws    = (float*)d_ws;
  float* Mbase = ws;
  float* P0    = ws + 2 * HH * HH;
  float* P1    = ws + 4 * HH * HH;
  float* F     = ws + 6 * HH * HH;
  float* Cs    = F + NC * 4096;

  build_M_kernel<<<64, 256, 0, stream>>>(Wh_re, Wh_im, Mbase);
  // M^2, M^4, M^8, M^16, M^32, M^64  -> final power in P1
  mat_square_kernel<<<8, 256, 0, stream>>>(Mbase, P0);
  mat_square_kernel<<<8, 256, 0, stream>>>(P0, P1);
  mat_square_kernel<<<8, 256, 0, stream>>>(P1, P0);
  mat_square_kernel<<<8, 256, 0, stream>>>(P0, P1);
  mat_square_kernel<<<8, 256, 0, stream>>>(P1, P0);
  mat_square_kernel<<<8, 256, 0, stream>>>(P0, P1);

  const size_t smChunk1 = (size_t)(2 * HH * BSTR + 2 * DIN * BSTR + 4 * BB * ASTR) * sizeof(float);
  const size_t smChunk3 = smChunk1 + (size_t)(2 * HH * OSTR) * sizeof(float);
  const size_t smCarry  = (size_t)(2 * HH * BSTR + 4 * BB * ASTR) * sizeof(float);

  chunk_kernel<false><<<NC, 256, smChunk1, stream>>>(
      x_re, x_im, Wi_re, Wi_im, nullptr, nullptr, Mbase, nullptr, F, nullptr);
  carry_kernel<<<1, 256, smCarry, stream>>>(P1, F, Cs);
  chunk_kernel<true><<<NC, 256, smChunk3, stream>>>(
      x_re, x_im, Wi_re, Wi_im, Wo_re, Wo_im, Mbase, Cs, nullptr, out);
}